// GNNReasoner_1202590842941
// MI455X (gfx1250) — compile-verified
//
#include <hip/hip_runtime.h>

#define NEG_SLOPE 0.2f
#define LN_EPS 1e-5f
#define SM_EPS 1e-16f

typedef __attribute__((ext_vector_type(2))) float v2f;
typedef __attribute__((ext_vector_type(8))) float v8f;

__device__ __forceinline__ float lrelu(float x) { return x > 0.0f ? x : NEG_SLOPE * x; }

// ---------------------------------------------------------------------------
// C[nrows,64] = A[nrows,64] * W[64,64] using fp32 WMMA 16x16x4.
// Block = 128 threads (4 waves). Each block handles 16 rows; wave w handles
// the 16-column strip [16w, 16w+16). K loop: 16 steps of K=4.
// A frag (16x4 f32): lanes 0-15 hold M=lane, VGPR0/1 = K0/K1;
//                    lanes 16-31 hold M=lane-16, VGPR0/1 = K2/K3.
// B frag (4x16 f32): lanes 0-15 hold N=lane, VGPR0/1 = K0/K1 rows;
//                    lanes 16-31 VGPR0/1 = K2/K3 rows.
// D (16x16 f32): VGPR r, lane l -> row r + 8*(l>>4), col l&15.
// OOB rows are CLAMPED (branch-free; EXEC stays all-1s for WMMA) — their
// results are simply not stored.
// ---------------------------------------------------------------------------
__global__ void __launch_bounds__(128)
gemm64_wmma(const float* __restrict__ A, const float* __restrict__ W,
            float* __restrict__ C, int nrows) {
    const int wave = threadIdx.x >> 5;
    const int lane = threadIdx.x & 31;
    const int mrow = lane & 15;   // row in tile (A) / col in strip (B)
    const int kh   = lane >> 4;   // k-pair selector (0 -> K0/K1, 1 -> K2/K3)
    const int row0 = blockIdx.x * 16;
    int row = row0 + mrow;
    row = row < nrows ? row : nrows - 1;          // clamp: no EXEC divergence
    const int col  = wave * 16 + mrow;

    const float* ap = A + (size_t)row * 64 + 2 * kh;
    const float* bp = W + (size_t)(2 * kh) * 64 + col;

    v8f acc = {};
    #pragma unroll
    for (int k = 0; k < 64; k += 4) {
        v2f a = *(const v2f*)(ap + k);            // global_load_b64
        v2f b;
        b.x = bp[(size_t)k * 64];
        b.y = bp[(size_t)k * 64 + 64];
        acc = __builtin_amdgcn_wmma_f32_16x16x4_f32(
            /*neg_a=*/false, a, /*neg_b=*/false, b,
            /*c_mod=*/(short)0, acc, /*reuse_a=*/false, /*reuse_b=*/false);
    }

    const int srow = row0 + 8 * (lane >> 4);
    const int scol = wave * 16 + (lane & 15);
    #pragma unroll
    for (int r = 0; r < 8; ++r) {
        const int rr = srow + r;
        if (rr < nrows) C[(size_t)rr * 64 + scol] = acc[r];
    }
}

// ---------------------------------------------------------------------------
// Per (node, head): attention dots + self-loop max init.
// ---------------------------------------------------------------------------
template <int H, int C>
__global__ void attn_prep(const float* __restrict__ xw,
                          const float* __restrict__ a_src,
                          const float* __restrict__ a_dst,
                          float* __restrict__ asrc, float* __restrict__ adst,
                          float* __restrict__ m, int n) {
    const int t = blockIdx.x * blockDim.x + threadIdx.x;
    if (t >= n * H) return;
    const int node = t / H;
    const int h    = t - node * H;
    const float* xp = xw + (size_t)node * 64 + h * C;
    float s = 0.0f, d = 0.0f;
    #pragma unroll
    for (int c = 0; c < C; ++c) {
        const float v = xp[c];
        s += v * a_src[h * C + c];
        d += v * a_dst[h * C + c];
    }
    asrc[t] = s;
    adst[t] = d;
    m[t] = lrelu(s + d);   // self-loop edge participates in the segment max
}

// ---------------------------------------------------------------------------
// Per (edge, head): segment max of leaky_relu(asrc[s]+adst[d]) into m[dst].
// ---------------------------------------------------------------------------
template <int H>
__global__ void edge_max_k(const int* __restrict__ src, const int* __restrict__ dst,
                           const float* __restrict__ asrc, const float* __restrict__ adst,
                           float* __restrict__ m, int ecnt) {
    const int t = blockIdx.x * blockDim.x + threadIdx.x;
    if (t >= ecnt * H) return;
    const int edge = t / H;
    const int h    = t - edge * H;
    const int s = src[edge];
    const int d = dst[edge];
    const float e = lrelu(asrc[s * H + h] + adst[d * H + h]);
    atomicMax(&m[d * H + h], e);
}

// ---------------------------------------------------------------------------
// Per (node, channel): self-loop term initializes denom and agg (plain
// stores, so no pre-zeroing of the accumulators is required).
// ---------------------------------------------------------------------------
template <int H, int C>
__global__ void softmax_self(const float* __restrict__ xw,
                             const float* __restrict__ asrc,
                             const float* __restrict__ adst,
                             const float* __restrict__ m,
                             float* __restrict__ denom, float* __restrict__ agg,
                             int n) {
    const int t = blockIdx.x * blockDim.x + threadIdx.x;
    if (t >= n * 64) return;
    const int node = t >> 6;
    const int ch   = t & 63;
    const int h    = ch / C;
    const float e  = lrelu(asrc[node * H + h] + adst[node * H + h]);
    const float ex = __expf(e - m[node * H + h]);
    agg[t] = xw[t] * ex;
    if ((ch & (C - 1)) == 0) denom[node * H + h] = ex;
}

// ---------------------------------------------------------------------------
// Per (edge, channel): ex = exp(e - m[dst]); agg[dst] += ex * xw[src];
// one lane per head also accumulates denom. 64 lanes cooperate per edge.
// Unnormalized accumulation: final division by (denom+eps) happens per node
// in finalize_k (same result as per-edge normalization, half the passes).
// ---------------------------------------------------------------------------
template <int H, int C>
__global__ void edge_agg(const int* __restrict__ src, const int* __restrict__ dst,
                         const float* __restrict__ asrc, const float* __restrict__ adst,
                         const float* __restrict__ m, const float* __restrict__ xw,
                         float* __restrict__ denom, float* __restrict__ agg,
                         int ecnt) {
    const long long t = (long long)blockIdx.x * blockDim.x + threadIdx.x;
    if (t >= (long long)ecnt * 64) return;
    const int edge = (int)(t >> 6);
    const int ch   = (int)(t & 63);
    const int h    = ch / C;
    const int s = src[edge];
    const int d = dst[edge];
    const float e  = lrelu(asrc[s * H + h] + adst[d * H + h]);
    const float ex = __expf(e - m[d * H + h]);
    atomicAdd(&agg[(size_t)d * 64 + ch], ex * xw[(size_t)s * 64 + ch]);
    if ((ch & (C - 1)) == 0) atomicAdd(&denom[d * H + h], ex);
}

// ---------------------------------------------------------------------------
// Per node (one wave32, 2 channels/lane): normalize by denom, add bias,
// LayerNorm (shuffle reductions), ELU.
// ---------------------------------------------------------------------------
template <int H, int C>
__global__ void finalize_k(const float* __restrict__ agg, const float* __restrict__ denom,
                           const float* __restrict__ bias, const float* __restrict__ g,
                           const float* __restrict__ bparam, float* __restrict__ out,
                           int n) {
    const int node = (blockIdx.x * blockDim.x + threadIdx.x) >> 5;
    const int lane = threadIdx.x & 31;
    if (node >= n) return;
    const int ch0 = lane, ch1 = lane + 32;
    float v0 = agg[(size_t)node * 64 + ch0] / (denom[node * H + ch0 / C] + SM_EPS) + bias[ch0];
    float v1 = agg[(size_t)node * 64 + ch1] / (denom[node * H + ch1 / C] + SM_EPS) + bias[ch1];

    float s = v0 + v1;
    #pragma unroll
    for (int off = 16; off > 0; off >>= 1) s += __shfl_xor(s, off, 32);
    const float mu = s * (1.0f / 64.0f);
    const float d0 = v0 - mu, d1 = v1 - mu;
    float vv = d0 * d0 + d1 * d1;
    #pragma unroll
    for (int off = 16; off > 0; off >>= 1) vv += __shfl_xor(vv, off, 32);
    const float rstd = rsqrtf(vv * (1.0f / 64.0f) + LN_EPS);

    float o0 = d0 * rstd * g[ch0] + bparam[ch0];
    float o1 = d1 * rstd * g[ch1] + bparam[ch1];
    o0 = o0 > 0.0f ? o0 : __expf(o0) - 1.0f;   // ELU
    o1 = o1 > 0.0f ? o1 : __expf(o1) - 1.0f;
    out[(size_t)node * 64 + ch0] = o0;
    out[(size_t)node * 64 + ch1] = o1;
}

static inline unsigned cdiv(long long a, int b) { return (unsigned)((a + b - 1) / b); }

extern "C" void kernel_launch(void* const* d_in, const int* in_sizes, int n_in,
                              void* d_out, int out_size, void* d_ws, size_t ws_size,
                              hipStream_t stream) {
    const float* h   = (const float*)d_in[0];
    const int*   ei  = (const int*)d_in[1];
    const float* W1  = (const float*)d_in[2];
    const float* a1s = (const float*)d_in[3];
    const float* a1d = (const float*)d_in[4];
    const float* b1  = (const float*)d_in[5];
    const float* W2  = (const float*)d_in[6];
    const float* a2s = (const float*)d_in[7];
    const float* a2d = (const float*)d_in[8];
    const float* b2  = (const float*)d_in[9];
    const float* ln1g = (const float*)d_in[10];
    const float* ln1b = (const float*)d_in[11];
    const float* ln2g = (const float*)d_in[12];
    const float* ln2b = (const float*)d_in[13];

    const int n = in_sizes[0] / 64;
    const int e = in_sizes[1] / 2;
    const int* src = ei;
    const int* dst = ei + e;

    float* ws = (float*)d_ws;
    float* xw   = ws; ws += (size_t)n * 64;
    float* asrc = ws; ws += (size_t)n * 4;
    float* adst = ws; ws += (size_t)n * 4;
    float* m    = ws; ws += (size_t)n * 4;
    float* den  = ws; ws += (size_t)n * 4;
    float* agg  = ws; ws += (size_t)n * 64;
    float* x1   = ws; ws += (size_t)n * 64;

    const int B = 256;

    // ---------------- Layer 1: H=4, C=16, concat ----------------
    gemm64_wmma<<<cdiv(n, 16), 128, 0, stream>>>(h, W1, xw, n);
    attn_prep<4, 16><<<cdiv((long long)n * 4, B), B, 0, stream>>>(xw, a1s, a1d, asrc, adst, m, n);
    edge_max_k<4><<<cdiv((long long)e * 4, B), B, 0, stream>>>(src, dst, asrc, adst, m, e);
    softmax_self<4, 16><<<cdiv((long long)n * 64, B), B, 0, stream>>>(xw, asrc, adst, m, den, agg, n);
    edge_agg<4, 16><<<cdiv((long long)e * 64, B), B, 0, stream>>>(src, dst, asrc, adst, m, xw, den, agg, e);
    finalize_k<4, 16><<<cdiv((long long)n * 32, B), B, 0, stream>>>(agg, den, b1, ln1g, ln1b, x1, n);

    // ---------------- Layer 2: H=1, C=64, mean (identity) -------
    gemm64_wmma<<<cdiv(n, 16), 128, 0, stream>>>(x1, W2, xw, n);
    attn_prep<1, 64><<<cdiv((long long)n, B), B, 0, stream>>>(xw, a2s, a2d, asrc, adst, m, n);
    edge_max_k<1><<<cdiv((long long)e, B), B, 0, stream>>>(src, dst, asrc, adst, m, e);
    softmax_self<1, 64><<<cdiv((long long)n * 64, B), B, 0, stream>>>(xw, asrc, adst, m, den, agg, n);
    edge_agg<1, 64><<<cdiv((long long)e * 64, B), B, 0, stream>>>(src, dst, asrc, adst, m, xw, den, agg, e);
    finalize_k<1, 64><<<cdiv((long long)n * 32, B), B, 0, stream>>>(agg, den, b2, ln2g, ln2b, (float*)d_out, n);
}